// NFFT_32504312496637
// MI455X (gfx1250) — compile-verified
//
#include <hip/hip_runtime.h>
#include <math.h>

// ---------------- CDNA5 vector types for WMMA ----------------
typedef __attribute__((ext_vector_type(2))) float v2f;  // 16x4 / 4x16 f32 fragment (2 VGPRs)
typedef __attribute__((ext_vector_type(8))) float v8f;  // 16x16 f32 C/D fragment (8 VGPRs)

#define NFFT_N     16384
#define NFFT_NOVR  32768
#define NTHREADS   1024
#define PI_F       3.14159265358979f
#define TWO_PI_F   6.28318530717959f
#define B_KB       (1.5f * PI_F)          // (2 - 1/sigma)*pi, sigma = 2

// ---------------- Kaiser-Bessel helpers ----------------
// I0(x) for x >= 3.75 (Abramowitz & Stegun 9.8.2). Here x in [17.8, 18.9].
__device__ __forceinline__ float i0_large(float x) {
    float t = 3.75f / x;
    float p = 0.39894228f + t * (0.01328592f + t * (0.00225319f + t * (-0.00157565f +
              t * (0.00916281f + t * (-0.02057706f + t * (0.02635537f +
              t * (-0.01647633f + t * 0.00392377f)))))));
    return __expf(x) * p * rsqrtf(x);
}

// phi(k) evaluated with nk = n*x - idx  (so k = nk/n); includes /SCALE.
__device__ __forceinline__ float kb_window(float nk) {
    float t = 16.0f - nk * nk;          // m^2 - (n k)^2, m = 4
    if (fabsf(nk) > 4.0f) return 0.0f;
    float w;
    if (t > 0.0f) {
        float a = sqrtf(t);
        float z = B_KB * a;
        float sh = 0.5f * (__expf(z) - __expf(-z));
        w = sh / (a * PI_F);
    } else {
        w = B_KB / PI_F;                // boundary |k| == m/n
    }
    return w * 1e-10f;
}

// ---------------- WMMA radix-16 DIF stage ----------------
// One wave processes one group of 16 butterflies as a complex 16x16 matmul:
//   Y = DFT16 * X  via 16 chained V_WMMA_F32_16X16X4_F32 (K accumulated in 4 chunks).
// In-place in LDS; twiddle W_{16S}^{j*b} applied on writeback (positive exponent / IDFT).
template<int LOG2S>
__device__ __forceinline__ void radix16_stage(float* __restrict__ sre,
                                              float* __restrict__ sim,
                                              const v2f* Dr, const v2f* Di, const v2f* nDi,
                                              int wave, int lane) {
    constexpr int S = 1 << LOG2S;
    const int q    = lane & 15;   // column (butterfly within group) / N index
    const int half = lane >> 4;
    const float twq = TWO_PI_F / (float)(S * 16);

    for (int g = wave; g < 128; g += 32) {        // 2048 butterflies / 16 per group
        int t       = g * 16 + q;                 // global butterfly id for this column
        int blk     = t >> LOG2S;
        int jj      = t & (S - 1);
        int colbase = (blk << (LOG2S + 4)) + jj;  // element a lives at colbase + S*a

        // Load B fragments (4x16 data chunks): lane holds K = 4c + 2*half + {0,1}
        v2f Xr[4], Xi[4];
#pragma unroll
        for (int c = 0; c < 4; ++c) {
            int a0 = 4 * c + 2 * half;
            int i0a = colbase + S * a0;
            Xr[c][0] = sre[i0a];     Xr[c][1] = sre[i0a + S];
            Xi[c][0] = sim[i0a];     Xi[c][1] = sim[i0a + S];
        }

        // Complex matmul: Yr = Dr*Xr - Di*Xi ; Yi = Di*Xr + Dr*Xi
        v8f Yr = {}; v8f Yi = {};
#pragma unroll
        for (int c = 0; c < 4; ++c)
            Yr = __builtin_amdgcn_wmma_f32_16x16x4_f32(false, Dr[c],  false, Xr[c], (short)0, Yr, false, false);
#pragma unroll
        for (int c = 0; c < 4; ++c)
            Yr = __builtin_amdgcn_wmma_f32_16x16x4_f32(false, nDi[c], false, Xi[c], (short)0, Yr, false, false);
#pragma unroll
        for (int c = 0; c < 4; ++c)
            Yi = __builtin_amdgcn_wmma_f32_16x16x4_f32(false, Di[c],  false, Xr[c], (short)0, Yi, false, false);
#pragma unroll
        for (int c = 0; c < 4; ++c)
            Yi = __builtin_amdgcn_wmma_f32_16x16x4_f32(false, Dr[c],  false, Xi[c], (short)0, Yi, false, false);

        // Twiddle + in-place writeback. D layout: VGPR r holds row M = r + 8*half.
        float jf = (float)jj;
#pragma unroll
        for (int r = 0; r < 8; ++r) {
            int bI = r + 8 * half;
            float sn, cs;
            __sincosf(twq * jf * (float)bI, &sn, &cs);
            float yr = Yr[r], yi = Yi[r];
            int addr = colbase + S * bI;
            sre[addr] = yr * cs - yi * sn;
            sim[addr] = yr * sn + yi * cs;
        }
    }
}

// ---------------- Fused NFFT kernel: one workgroup per batch row ----------------
// LDS (256 KB of the WGP's 320 KB) holds the full 32768-pt complex signal:
//   deconvolve -> 3x WMMA radix-16 stages -> radix-8 -> digit-reversed gather.
__global__ __launch_bounds__(NTHREADS)
void nfft_fused_kernel(const float* __restrict__ x,
                       const float* __restrict__ f_re,
                       const float* __restrict__ f_im,
                       float* __restrict__ out) {
    extern __shared__ float smem[];
    float* sre = smem;
    float* sim = smem + NFFT_NOVR;

    const int tid  = threadIdx.x;
    const int b    = blockIdx.x;
    const int lane = tid & 31;
    const int wave = tid >> 5;

    // ---- Phase 1: deconvolve by phi_hat, place in natural FFT order (zero-padded) ----
    {
        const float* fre = f_re + (size_t)b * NFFT_N;
        const float* fim = f_im + (size_t)b * NFFT_N;
        const float bb = B_KB * B_KB;                       // 22.2066
        for (int i = tid; i < NFFT_N; i += NTHREADS) {
            float f   = (float)(i - NFFT_N / 2);            // freq in [-8192, 8192)
            float wf  = f * (PI_F / 16384.0f);              // 2*pi*f/n
            float arg = 4.0f * sqrtf(bb - wf * wf);
            float ph  = i0_large(arg) * 1e-10f;             // phi_hat
            float inv = 1.0f / ph;
            int k = (i >= NFFT_N / 2) ? (i - NFFT_N / 2) : (i + 24576);
            sre[k] = fre[i] * inv;
            sim[k] = fim[i] * inv;
        }
        for (int k = NFFT_N / 2 + tid; k < 24576; k += NTHREADS) {
            sre[k] = 0.0f; sim[k] = 0.0f;
        }
    }
    __syncthreads();

    // ---- Per-lane DFT16 matrix fragments (positive exponent): D[M][K] = e^{+2pi i M K/16}
    v2f Dr[4], Di[4], nDi[4];
    {
        const int half = lane >> 4;
        const int Mrow = lane & 15;
#pragma unroll
        for (int c = 0; c < 4; ++c) {
#pragma unroll
            for (int j = 0; j < 2; ++j) {
                int K = 4 * c + 2 * half + j;
                float sn, cs;
                __sincosf(0.392699081698724f * (float)(Mrow * K), &sn, &cs); // 2pi/16
                Dr[c][j] = cs; Di[c][j] = sn; nDi[c][j] = -sn;
            }
        }
    }

    // ---- Phase 2: DIF FFT, radices 16,16,16 (WMMA) then 8 (VALU); output digit-reversed
    radix16_stage<11>(sre, sim, Dr, Di, nDi, wave, lane);   // S = 2048
    __syncthreads();
    radix16_stage<7>(sre, sim, Dr, Di, nDi, wave, lane);    // S = 128
    __syncthreads();
    radix16_stage<3>(sre, sim, Dr, Di, nDi, wave, lane);    // S = 8
    __syncthreads();

    // Radix-8 final stage: 8-pt positive-exponent DFT on contiguous octets, in place.
    for (int blk8 = tid; blk8 < NFFT_NOVR / 8; blk8 += NTHREADS) {
        int base = blk8 * 8;
        float zr[8], zi[8];
#pragma unroll
        for (int a = 0; a < 8; ++a) { zr[a] = sre[base + a]; zi[a] = sim[base + a]; }
        // E = DFT4(z0,z2,z4,z6), O = DFT4(z1,z3,z5,z7)   (Y1 = t1 + i t3, Y3 = t1 - i t3)
        float t0r = zr[0] + zr[4], t0i = zi[0] + zi[4];
        float t1r = zr[0] - zr[4], t1i = zi[0] - zi[4];
        float t2r = zr[2] + zr[6], t2i = zi[2] + zi[6];
        float t3r = zr[2] - zr[6], t3i = zi[2] - zi[6];
        float E0r = t0r + t2r, E0i = t0i + t2i;
        float E2r = t0r - t2r, E2i = t0i - t2i;
        float E1r = t1r - t3i, E1i = t1i + t3r;
        float E3r = t1r + t3i, E3i = t1i - t3r;
        float u0r = zr[1] + zr[5], u0i = zi[1] + zi[5];
        float u1r = zr[1] - zr[5], u1i = zi[1] - zi[5];
        float u2r = zr[3] + zr[7], u2i = zi[3] + zi[7];
        float u3r = zr[3] - zr[7], u3i = zi[3] - zi[7];
        float O0r = u0r + u2r, O0i = u0i + u2i;
        float O2r = u0r - u2r, O2i = u0i - u2i;
        float O1r = u1r - u3i, O1i = u1i + u3r;
        float O3r = u1r + u3i, O3i = u1i - u3r;
        // Twiddle O by W8^b = e^{+2pi i b/8}
        const float rr = 0.70710678118655f;
        float w1r = rr * (O1r - O1i), w1i = rr * (O1r + O1i);   // * (r + i r)
        float w2r = -O2i,             w2i = O2r;                // * i
        float w3r = -rr * (O3r + O3i), w3i = rr * (O3r - O3i);  // * (-r + i r)
        sre[base + 0] = E0r + O0r;  sim[base + 0] = E0i + O0i;
        sre[base + 4] = E0r - O0r;  sim[base + 4] = E0i - O0i;
        sre[base + 1] = E1r + w1r;  sim[base + 1] = E1i + w1i;
        sre[base + 5] = E1r - w1r;  sim[base + 5] = E1i - w1i;
        sre[base + 2] = E2r + w2r;  sim[base + 2] = E2i + w2i;
        sre[base + 6] = E2r - w2r;  sim[base + 6] = E2i - w2i;
        sre[base + 3] = E3r + w3r;  sim[base + 3] = E3i + w3i;
        sre[base + 7] = E3r - w3r;  sim[base + 7] = E3i - w3i;
    }
    __syncthreads();

    // ---- Phase 3: windowed gather straight from digit-reversed LDS (no reorder pass)
    // g[(idx+16384) wrapped] collapses to G[idx mod n]; G[t] lives at digitrev(t).
    const float* xrow = x + (size_t)b * NFFT_NOVR;
    float* orow = out + (size_t)b * (size_t)NFFT_NOVR * 2;
    for (int p = tid; p < NFFT_NOVR; p += NTHREADS) {
        float xv  = xrow[p];
        float nx  = (float)NFFT_NOVR * xv;
        int  idx0 = (int)ceilf(nx) - 4;
        float accr = 0.0f, acci = 0.0f;
#pragma unroll
        for (int l = 0; l < 8; ++l) {
            int idx  = idx0 + l;
            float w  = kb_window(nx - (float)idx);
            int tau  = (idx + NFFT_NOVR) & (NFFT_NOVR - 1);
            int pos  = ((tau & 15) << 11) | (((tau >> 4) & 15) << 7)
                     | (((tau >> 8) & 15) << 3) | (tau >> 12);
            accr = fmaf(w, sre[pos], accr);
            acci = fmaf(w, sim[pos], acci);
        }
        orow[2 * p]     = accr;   // complex64 interleaved
        orow[2 * p + 1] = acci;
    }
}

// ---------------- Host launch ----------------
extern "C" void kernel_launch(void* const* d_in, const int* in_sizes, int n_in,
                              void* d_out, int out_size, void* d_ws, size_t ws_size,
                              hipStream_t stream) {
    (void)in_sizes; (void)n_in; (void)out_size; (void)d_ws; (void)ws_size;
    const float* x    = (const float*)d_in[0];   // (64, 32768)
    const float* f_re = (const float*)d_in[1];   // (64, 16384)
    const float* f_im = (const float*)d_in[2];   // (64, 16384)
    float* out        = (float*)d_out;           // (64, 32768) complex64 interleaved

    const int ldsBytes = 2 * NFFT_NOVR * (int)sizeof(float);   // 256 KB of 320 KB WGP LDS
    (void)hipFuncSetAttribute((const void*)nfft_fused_kernel,
                              hipFuncAttributeMaxDynamicSharedMemorySize, ldsBytes);
    nfft_fused_kernel<<<64, NTHREADS, ldsBytes, stream>>>(x, f_re, f_im, out);
}